// SimpleSSM_9766755631856
// MI455X (gfx1250) — compile-verified
//
#include <hip/hip_runtime.h>
#include <stdint.h>

// Problem geometry (fixed by the reference).
#define T        4096      // time steps
#define DIMV     256       // dim/4 float4 channels (dim = 1024)
#define TCHUNK   64        // time steps per chunk
#define NCHUNK   (T / TCHUNK)   // 64 chunks
#define TT       8         // time steps per LDS tile in kernel A
#define NTILE    (TCHUNK / TT)  // 8 tiles per chunk
#define BDIM     256       // threads per block (= DIMV)

typedef float vfloat4 __attribute__((ext_vector_type(4)));   // native vector for NT store

__device__ __forceinline__ float4 f4mul(float4 a, float4 b) {
    return make_float4(a.x * b.x, a.y * b.y, a.z * b.z, a.w * b.w);
}
__device__ __forceinline__ float4 f4fma(float4 a, float4 b, float4 c) {
    // a*b + c, componentwise
    return make_float4(fmaf(a.x, b.x, c.x), fmaf(a.y, b.y, c.y),
                       fmaf(a.z, b.z, c.z), fmaf(a.w, b.w, c.w));
}
__device__ __forceinline__ void nt_store_f4(float4 v, float4* p) {
    vfloat4 t;
    t.x = v.x; t.y = v.y; t.z = v.z; t.w = v.w;
    __builtin_nontemporal_store(t, (vfloat4*)p);
}

// Issue async global->LDS copies for one tile (TT*DIMV float4 = 32KB),
// BDIM threads, (TT*DIMV)/BDIM = 8 x b128 per thread. CDNA5 async path:
// tracked by ASYNCcnt, data lands directly in LDS without touching VGPRs.
__device__ __forceinline__ void async_copy_tile(const float4* __restrict__ gsrc,
                                                float4* lds_dst, int tid) {
#pragma unroll
    for (int i = 0; i < (TT * DIMV) / BDIM; ++i) {
        int idx = tid + i * BDIM;
        // Low 32 bits of a generic pointer into LDS == LDS byte offset
        // (ISA 10.2: LDS aperture address truncates to LDS_ADDR.U32).
        uint32_t lds_off = (uint32_t)(uintptr_t)(lds_dst + idx);
        unsigned long long gaddr = (unsigned long long)(uintptr_t)(gsrc + idx);
        asm volatile("global_load_async_to_lds_b128 %0, %1, off"
                     :: "v"(lds_off), "v"(gaddr)
                     : "memory");
    }
}

// ---------------------------------------------------------------------------
// Kernel A: per-chunk local scan (zero entry state) -> chunk summary u.
// Grid: (NCHUNK, batch). Block: 256 threads = 256 float4 channels.
// x staged through LDS with double-buffered async copies.
// ---------------------------------------------------------------------------
__global__ __launch_bounds__(BDIM)
void ssm_chunk_summary(const float4* __restrict__ x,
                       const float4* __restrict__ A,
                       const float4* __restrict__ B,
                       float4* __restrict__ u) {
    __shared__ float4 buf[2][TT * DIMV];   // 2 x 32 KB

    const int dv = threadIdx.x;            // float4 channel
    const int j  = blockIdx.x;             // chunk
    const int b  = blockIdx.y;             // batch
    const long base = ((long)b * T + (long)j * TCHUNK) * DIMV; // float4 idx

    const float4 a  = A[dv];
    const float4 bb = B[dv];
    float4 s = make_float4(0.f, 0.f, 0.f, 0.f);

    // Prologue: tile 0 in flight.
    async_copy_tile(x + base, &buf[0][0], dv);

    for (int k = 0; k < NTILE; ++k) {
        if (k + 1 < NTILE) {
            async_copy_tile(x + base + (long)(k + 1) * TT * DIMV,
                            &buf[(k + 1) & 1][0], dv);
            // 8 newly issued outstanding; wait until tile k's 8 are done
            // (async loads retire in order).
            asm volatile("s_wait_asynccnt 0x8" ::: "memory");
        } else {
            asm volatile("s_wait_asynccnt 0x0" ::: "memory");
        }
        __syncthreads();   // all waves' tile-k data visible in LDS
#pragma unroll
        for (int t = 0; t < TT; ++t) {
            float4 xv = buf[k & 1][t * DIMV + dv];
            s = f4fma(a, s, f4mul(bb, xv));     // s = a*s + b*x
        }
        __syncthreads();   // done reading buf[k&1] before it is refilled
    }

    u[((long)b * NCHUNK + j) * DIMV + dv] = s;  // 2 MB total: keeps L2 warm with x
}

// ---------------------------------------------------------------------------
// Kernel B: exclusive scan over chunk summaries (in place): u[j] becomes the
// entry state of chunk j.  c_0 = 0 ; c_{j+1} = a^64 * c_j + u_j.
// Grid: (batch). Block: 256 threads.
// ---------------------------------------------------------------------------
__global__ __launch_bounds__(BDIM)
void ssm_chunk_scan(const float4* __restrict__ A, float4* __restrict__ u) {
    const int dv = threadIdx.x;
    const int b  = blockIdx.x;

    float4 p = A[dv];
#pragma unroll
    for (int i = 0; i < 6; ++i) p = f4mul(p, p);   // a^64

    float4 c = make_float4(0.f, 0.f, 0.f, 0.f);
    const long base = (long)b * NCHUNK * DIMV + dv;
    for (int j = 0; j < NCHUNK; ++j) {
        float4 uj = u[base + (long)j * DIMV];
        u[base + (long)j * DIMV] = c;
        c = f4fma(p, c, uj);
    }
}

// ---------------------------------------------------------------------------
// Kernel C: replay the exact recurrence per chunk seeded with the entry state,
// writing every output. x re-read (expected L2 hits: 128 MB fits in 192 MB L2),
// outputs written with non-temporal stores to avoid evicting x mid-kernel.
// Grid: (NCHUNK, batch). Block: 256 threads.
// ---------------------------------------------------------------------------
__global__ __launch_bounds__(BDIM)
void ssm_final(const float4* __restrict__ x,
               const float4* __restrict__ A,
               const float4* __restrict__ B,
               const float4* __restrict__ carry,
               float4* __restrict__ out) {
    const int dv = threadIdx.x;
    const int j  = blockIdx.x;
    const int b  = blockIdx.y;

    const float4 a  = A[dv];
    const float4 bb = B[dv];
    float4 s = carry[((long)b * NCHUNK + j) * DIMV + dv];

    const long base = ((long)b * T + (long)j * TCHUNK) * DIMV + dv;
#pragma unroll 4
    for (int t = 0; t < TCHUNK; ++t) {
        float4 xv = x[base + (long)t * DIMV];
        s = f4fma(a, s, f4mul(bb, xv));
        nt_store_f4(s, &out[base + (long)t * DIMV]);
    }
}

// ---------------------------------------------------------------------------
extern "C" void kernel_launch(void* const* d_in, const int* in_sizes, int n_in,
                              void* d_out, int out_size, void* d_ws, size_t ws_size,
                              hipStream_t stream) {
    const float4* x = (const float4*)d_in[0];   // [batch, 4096, 1024] f32
    const float4* A = (const float4*)d_in[1];   // [1024] f32
    const float4* B = (const float4*)d_in[2];   // [1024] f32
    float4* out = (float4*)d_out;

    const int batch = in_sizes[0] / (T * DIMV * 4);   // = 8

    float4* u = (float4*)d_ws;   // batch * NCHUNK * DIMV float4 = 2 MB

    dim3 blk(BDIM);
    ssm_chunk_summary<<<dim3(NCHUNK, batch), blk, 0, stream>>>(x, A, B, u);
    ssm_chunk_scan   <<<dim3(batch),          blk, 0, stream>>>(A, u);
    ssm_final        <<<dim3(NCHUNK, batch),  blk, 0, stream>>>(x, A, B, u, out);
}